// OptimizedGNN_83021717832548
// MI455X (gfx1250) — compile-verified
//
#include <hip/hip_runtime.h>
#include <math.h>

// ---------------- constants from the reference ----------------
#define N_NODESC 100000
#define N_EDGESC 1600000
#define N_GRAPHSC 1024
#define D_INC 100
#define HC 128
#define OUTC 2
#define EPSC 1e-5f

typedef __attribute__((ext_vector_type(2))) float v2f;
typedef __attribute__((ext_vector_type(8))) float v8f;
typedef unsigned int u32x4 __attribute__((ext_vector_type(4)));
typedef int i32x4 __attribute__((ext_vector_type(4)));
typedef int i32x8 __attribute__((ext_vector_type(8)));

// fp32 WMMA: D(16x16) = A(16x4) * B(4x16) + C   (exact fp32 path on CDNA5)
__device__ __forceinline__ v8f wmma_f32(v2f a, v2f b, v8f c) {
    return __builtin_amdgcn_wmma_f32_16x16x4_f32(
        /*neg_a=*/false, a, /*neg_b=*/false, b,
        /*c_mod=*/(short)0, c, /*reuse_a=*/false, /*reuse_b=*/false);
}

__device__ __forceinline__ float sigm(float x) { return 1.0f / (1.0f + expf(-x)); }

#if defined(__has_builtin)
#if __has_builtin(__builtin_amdgcn_tensor_load_to_lds)
#define HAVE_TDM 1
#endif
#endif

#ifdef HAVE_TDM
// Tensor Data Mover: async 2D tile load (f32 elements) global -> LDS.
// Descriptor packed per CDNA5 ISA D# spec (group0 + group1, 2D tensor,
// groups 2/3 disabled). Issued once per wave; tracked by TENSORcnt.
// This toolchain's builtin is the 6-arg form:
//   (u32x4 g0, i32x8 g1, i32x4 g2, i32x4 g3, i32x8 spare, i32 cpol)
__device__ __forceinline__ void tdm_load_2d_f32(void* lds_dst, const void* gsrc,
                                                unsigned tensor_d0, unsigned tensor_d1,
                                                unsigned tile_d0, unsigned tile_d1,
                                                unsigned stride0_elems) {
    const unsigned lds = (unsigned)(size_t)lds_dst;           // LDS byte offset
    const unsigned long long ga = (unsigned long long)(size_t)gsrc;
    u32x4 g0;
    g0[0] = 1u;                                               // count=1 (valid), user mode
    g0[1] = lds;                                              // lds_addr
    g0[2] = (unsigned)(ga & 0xFFFFFFFFu);                     // global_addr[31:0]
    g0[3] = (unsigned)((ga >> 32) & 0x01FFFFFFu) | (2u << 30);// global_addr[56:32] | type=2
    i32x8 g1;
    g1[0] = (int)(2u << 16);                                  // data_size=4B (code 2)
    g1[1] = (int)((tensor_d0 & 0xFFFFu) << 16);               // tensor_dim0[15:0]
    g1[2] = (int)((tensor_d0 >> 16) | ((tensor_d1 & 0xFFFFu) << 16));
    g1[3] = (int)((tensor_d1 >> 16) | ((tile_d0 & 0xFFFFu) << 16)); // tile_dim0
    g1[4] = (int)(tile_d1 & 0xFFFFu);                         // tile_dim1 ; tile_dim2=0
    g1[5] = (int)stride0_elems;                               // tensor_dim0_stride[31:0]
    g1[6] = 0;                                                // stride0 hi / stride1 lo
    g1[7] = 0;                                                // stride1 hi
    const i32x4 z4 = {0, 0, 0, 0};
    const i32x8 z8 = {0, 0, 0, 0, 0, 0, 0, 0};
    __builtin_amdgcn_tensor_load_to_lds(g0, g1, z4, z4, z8, 0);
}
#endif

// ---------------- zero fill (float4) ----------------
__global__ void k_zero4(float4* __restrict__ p, int n4) {
    int i = blockIdx.x * 256 + threadIdx.x;
    if (i < n4) p[i] = make_float4(0.f, 0.f, 0.f, 0.f);
}

// ---------------- small weight transpose: dst[c*K+k] = src[k*C+c] ----------------
__global__ void k_tr(const float* __restrict__ src, float* __restrict__ dst, int K, int C) {
    int i = blockIdx.x * 256 + threadIdx.x;
    if (i < K * C) {
        int k = i / C, c = i % C;
        dst[c * K + k] = src[i];
    }
}

// ---------------- input GEMM: h = relu(bn0(x @ w_in + b_in)) ----------------
// block = 256 (8 waves). block -> 16 rows; wave w -> cols [16w, 16w+16)
// w_inT is [H x D_IN] (transposed) so B fragments are one aligned v2f load.
__global__ void k_in_gemm(const float* __restrict__ x, const float* __restrict__ w_inT,
                          const float* __restrict__ bias,
                          const float* __restrict__ g, const float* __restrict__ b,
                          const float* __restrict__ mm, const float* __restrict__ vv,
                          float* __restrict__ h) {
    __shared__ __align__(16) float sx[16 * D_INC];
    const int tid = threadIdx.x;
    const int row0 = blockIdx.x * 16;
    {   // tile is a contiguous 1600-float span in global memory (full rows)
        const float4* x4 = (const float4*)(x + (size_t)row0 * D_INC);
        float4* s4 = (float4*)sx;
        for (int i = tid; i < 16 * D_INC / 4; i += 256) s4[i] = x4[i];
    }
    __builtin_prefetch(w_inT, 0, 3);
    __syncthreads();

    const int lane = tid & 31, m16 = lane & 15, hi = lane >> 4;
    const int col = (tid >> 5) * 16 + m16;
    const float* bp = w_inT + (size_t)col * D_INC;
    v8f acc = {};
    for (int kb = 0; kb < D_INC; kb += 4) {   // D_IN = 100 = 25*4 exactly
        const int k = kb + 2 * hi;
        v2f a; a.x = sx[m16 * D_INC + k]; a.y = sx[m16 * D_INC + k + 1];
        acc = wmma_f32(a, *(const v2f*)(bp + k), acc);
    }
    const float scale = g[col] * rsqrtf(vv[col] + EPSC);
    const float shift = (bias[col] - mm[col]) * scale + b[col];
#pragma unroll
    for (int r = 0; r < 8; ++r) {
        const int row = row0 + r + 8 * hi;
        h[(size_t)row * HC + col] = fmaxf(acc[r] * scale + shift, 0.0f);
    }
}

// ---------------- m = h @ W (HxH), WT is transposed [col][k] ----------------
__global__ void k_mm(const float* __restrict__ hin, const float* __restrict__ WT,
                     float* __restrict__ mout) {
    __shared__ __align__(16) float sh[16 * HC];
    const int tid = threadIdx.x;
    const int row0 = blockIdx.x * 16;
#ifdef HAVE_TDM
    if (tid < 32) {   // TDM 2D tile load: 16 rows x 128 f32, issued by wave 0
        tdm_load_2d_f32(sh, hin + (size_t)row0 * HC, HC, 16, HC, 16, HC);
        __builtin_amdgcn_s_wait_tensorcnt(0);
    }
#else
    {
        const float4* h4 = (const float4*)(hin + (size_t)row0 * HC);
        float4* s4 = (float4*)sh;
        for (int i = tid; i < 16 * HC / 4; i += 256) s4[i] = h4[i];
    }
#endif
    __builtin_prefetch(WT, 0, 3);
    __syncthreads();

    const int lane = tid & 31, m16 = lane & 15, hi = lane >> 4;
    const int col = (tid >> 5) * 16 + m16;
    const float* bp = WT + (size_t)col * HC;
    v8f acc = {};
    for (int kb = 0; kb < HC; kb += 4) {
        const int k = kb + 2 * hi;
        v2f a; a.x = sh[m16 * HC + k]; a.y = sh[m16 * HC + k + 1];
        acc = wmma_f32(a, *(const v2f*)(bp + k), acc);
    }
#pragma unroll
    for (int r = 0; r < 8; ++r)
        mout[(size_t)(row0 + r + 8 * hi) * HC + col] = acc[r];
}

// ---------------- scatter-add: agg[dst] += m[src]  (1 wave / edge) ----------------
__global__ void k_scatter(const float* __restrict__ m, const int* __restrict__ ei,
                          float* __restrict__ agg) {
    const int e = blockIdx.x * 8 + (threadIdx.x >> 5);
    const int lane = threadIdx.x & 31;
    if (e >= N_EDGESC) return;
    const int s = ei[e];
    const int d = ei[N_EDGESC + e];
    const float4 val = ((const float4*)(m + (size_t)s * HC))[lane];
    float* o = agg + (size_t)d * HC + lane * 4;
    atomicAdd(o + 0, val.x);
    atomicAdd(o + 1, val.y);
    atomicAdd(o + 2, val.z);
    atomicAdd(o + 3, val.w);
}

// ---------------- fused GRU: h = GRUCell(agg, h) -----------------
// gi = agg @ Wih^T + bih ; gh = h @ Whh^T + bhh ; gates in accumulator layout
__global__ void k_gru(float* __restrict__ h, const float* __restrict__ agg,
                      const float* __restrict__ wih, const float* __restrict__ whh,
                      const float* __restrict__ bih, const float* __restrict__ bhh) {
    __shared__ __align__(16) float s_agg[16 * HC];
    __shared__ __align__(16) float s_h[16 * HC];
    const int tid = threadIdx.x;
    const int row0 = blockIdx.x * 16;
#ifdef HAVE_TDM
    if (tid < 32) {   // two async TDM tile loads, then drain TENSORcnt
        tdm_load_2d_f32(s_agg, agg + (size_t)row0 * HC, HC, 16, HC, 16, HC);
        tdm_load_2d_f32(s_h, h + (size_t)row0 * HC, HC, 16, HC, 16, HC);
        __builtin_amdgcn_s_wait_tensorcnt(0);
    }
#else
    {
        const float4* a4 = (const float4*)(agg + (size_t)row0 * HC);
        const float4* h4 = (const float4*)(h + (size_t)row0 * HC);
        float4* sa4 = (float4*)s_agg;
        float4* sh4 = (float4*)s_h;
        for (int i = tid; i < 16 * HC / 4; i += 256) { sa4[i] = a4[i]; sh4[i] = h4[i]; }
    }
#endif
    __builtin_prefetch(wih, 0, 3);
    __builtin_prefetch(whh, 0, 3);
    __syncthreads();

    const int lane = tid & 31, m16 = lane & 15, hi = lane >> 4;
    const int col = (tid >> 5) * 16 + m16;
    // gi[n,j] = sum_k agg[n,k]*Wih[j,k] -> B[k][j] = Wih[j*H+k]: contiguous in k
    const float* wir = wih + (size_t)(0 * HC + col) * HC;
    const float* wiz = wih + (size_t)(1 * HC + col) * HC;
    const float* win = wih + (size_t)(2 * HC + col) * HC;
    const float* whr = whh + (size_t)(0 * HC + col) * HC;
    const float* whz = whh + (size_t)(1 * HC + col) * HC;
    const float* whn = whh + (size_t)(2 * HC + col) * HC;

    v8f ir = {}, iz = {}, inn = {}, hr = {}, hz = {}, hn = {};
    for (int kb = 0; kb < HC; kb += 4) {
        const int k = kb + 2 * hi;
        v2f aA; aA.x = s_agg[m16 * HC + k]; aA.y = s_agg[m16 * HC + k + 1];
        v2f aH; aH.x = s_h[m16 * HC + k];   aH.y = s_h[m16 * HC + k + 1];
        ir  = wmma_f32(aA, *(const v2f*)(wir + k), ir);
        iz  = wmma_f32(aA, *(const v2f*)(wiz + k), iz);
        inn = wmma_f32(aA, *(const v2f*)(win + k), inn);
        hr  = wmma_f32(aH, *(const v2f*)(whr + k), hr);
        hz  = wmma_f32(aH, *(const v2f*)(whz + k), hz);
        hn  = wmma_f32(aH, *(const v2f*)(whn + k), hn);
    }
    const float bir = bih[col], biz = bih[HC + col], bin = bih[2 * HC + col];
    const float bhr = bhh[col], bhz = bhh[HC + col], bhn = bhh[2 * HC + col];
#pragma unroll
    for (int r = 0; r < 8; ++r) {
        const int lrow = r + 8 * hi;
        const float rr   = sigm((ir[r] + bir) + (hr[r] + bhr));
        const float zz   = sigm((iz[r] + biz) + (hz[r] + bhz));
        const float cand = tanhf((inn[r] + bin) + rr * (hn[r] + bhn));
        const float hold = s_h[lrow * HC + col];
        h[(size_t)(row0 + lrow) * HC + col] = (1.0f - zz) * cand + zz * hold;
    }
}

// ---------------- bn1 + relu + global mean pool (accumulate) ----------------
__global__ void k_pool(const float* __restrict__ h, const int* __restrict__ batch,
                       const float* __restrict__ g, const float* __restrict__ b,
                       const float* __restrict__ mm, const float* __restrict__ vv,
                       float* __restrict__ pooled, float* __restrict__ counts) {
    const int node = blockIdx.x * 8 + (threadIdx.x >> 5);
    const int lane = threadIdx.x & 31;
    if (node >= N_NODESC) return;
    const int bg = batch[node];
    const float4 val = ((const float4*)(h + (size_t)node * HC))[lane];
    const int c0 = lane * 4;
    float tmp[4] = {val.x, val.y, val.z, val.w};
    float* p = pooled + (size_t)bg * HC + c0;
#pragma unroll
    for (int j = 0; j < 4; ++j) {
        const int c = c0 + j;
        const float scale = g[c] * rsqrtf(vv[c] + EPSC);
        const float t = (tmp[j] - mm[c]) * scale + b[c];
        atomicAdd(p + j, fmaxf(t, 0.0f));
    }
    if (lane == 0) atomicAdd(counts + bg, 1.0f);
}

// ---------------- fused head MLP on [G, H] (uses transposed fc1/fc2) ----------------
__global__ void k_head(const float* __restrict__ pooled, const float* __restrict__ counts,
                       const float* __restrict__ fc1T, const float* __restrict__ fc1b,
                       const float* __restrict__ g2, const float* __restrict__ b2,
                       const float* __restrict__ m2, const float* __restrict__ v2_,
                       const float* __restrict__ fc2T, const float* __restrict__ fc2b,
                       const float* __restrict__ g3, const float* __restrict__ b3,
                       const float* __restrict__ m3, const float* __restrict__ v3_,
                       const float* __restrict__ fc3w, const float* __restrict__ fc3b,
                       float* __restrict__ out) {
    __shared__ __align__(16) float s_g[16 * HC];
    __shared__ __align__(16) float s_g1[16 * HC];
    __shared__ __align__(16) float s_g2[16 * 64];
    const int tid = threadIdx.x;
    const int row0 = blockIdx.x * 16;
    for (int i = tid; i < 16 * HC; i += 256) {
        const int r = i >> 7, c = i & 127;
        const float cnt = fmaxf(counts[row0 + r], 1.0f);
        s_g[i] = pooled[(size_t)(row0 + r) * HC + c] / cnt;
    }
    __syncthreads();

    const int wave = tid >> 5, lane = tid & 31, m16 = lane & 15, hi = lane >> 4;
    {   // fc1 (128->128) + bn2 + relu
        const int col = wave * 16 + m16;
        const float* bp = fc1T + (size_t)col * HC;
        v8f acc = {};
        for (int kb = 0; kb < HC; kb += 4) {
            const int k = kb + 2 * hi;
            v2f a; a.x = s_g[m16 * HC + k]; a.y = s_g[m16 * HC + k + 1];
            acc = wmma_f32(a, *(const v2f*)(bp + k), acc);
        }
        const float scale = g2[col] * rsqrtf(v2_[col] + EPSC);
        const float shift = (fc1b[col] - m2[col]) * scale + b2[col];
#pragma unroll
        for (int r = 0; r < 8; ++r)
            s_g1[(r + 8 * hi) * HC + col] = fmaxf(acc[r] * scale + shift, 0.0f);
    }
    __syncthreads();
    if (wave < 4) {   // fc2 (128->64) + bn3 + relu
        const int col = wave * 16 + m16;
        const float* bp = fc2T + (size_t)col * HC;
        v8f acc = {};
        for (int kb = 0; kb < HC; kb += 4) {
            const int k = kb + 2 * hi;
            v2f a; a.x = s_g1[m16 * HC + k]; a.y = s_g1[m16 * HC + k + 1];
            acc = wmma_f32(a, *(const v2f*)(bp + k), acc);
        }
        const float scale = g3[col] * rsqrtf(v3_[col] + EPSC);
        const float shift = (fc2b[col] - m3[col]) * scale + b3[col];
#pragma unroll
        for (int r = 0; r < 8; ++r)
            s_g2[(r + 8 * hi) * 64 + col] = fmaxf(acc[r] * scale + shift, 0.0f);
    }
    __syncthreads();
    if (tid < 32) {   // fc3 (64->2)
        const int r = tid >> 1, o = tid & 1;
        float acc = fc3b[o];
        for (int k = 0; k < 64; ++k) acc += s_g2[r * 64 + k] * fc3w[k * 2 + o];
        out[(size_t)(row0 + r) * OUTC + o] = acc;
    }
}

// ---------------- host launcher ----------------
extern "C" void kernel_launch(void* const* d_in, const int* in_sizes, int n_in,
                              void* d_out, int out_size, void* d_ws, size_t ws_size,
                              hipStream_t stream) {
    (void)in_sizes; (void)n_in; (void)out_size; (void)ws_size;
    const float* x     = (const float*)d_in[0];
    const int*   ei    = (const int*)d_in[1];
    const int*   batch = (const int*)d_in[2];
    const float* w_in  = (const float*)d_in[3];
    const float* b_in  = (const float*)d_in[4];
    const float* bn0g = (const float*)d_in[5],  *bn0b = (const float*)d_in[6];
    const float* bn0m = (const float*)d_in[7],  *bn0v = (const float*)d_in[8];
    const float* ggc  = (const float*)d_in[9];
    const float* wih  = (const float*)d_in[10], *whh  = (const float*)d_in[11];
    const float* bih  = (const float*)d_in[12], *bhh  = (const float*)d_in[13];
    const float* bn1g = (const float*)d_in[14], *bn1b = (const float*)d_in[15];
    const float* bn1m = (const float*)d_in[16], *bn1v = (const float*)d_in[17];
    const float* fc1w = (const float*)d_in[18], *fc1b = (const float*)d_in[19];
    const float* bn2g = (const float*)d_in[20], *bn2b = (const float*)d_in[21];
    const float* bn2m = (const float*)d_in[22], *bn2v = (const float*)d_in[23];
    const float* fc2w = (const float*)d_in[24], *fc2b = (const float*)d_in[25];
    const float* bn3g = (const float*)d_in[26], *bn3b = (const float*)d_in[27];
    const float* bn3m = (const float*)d_in[28], *bn3v = (const float*)d_in[29];
    const float* fc3w = (const float*)d_in[30], *fc3b = (const float*)d_in[31];

    float* ws     = (float*)d_ws;
    float* h      = ws;                                  // N*H
    float* m      = h   + (size_t)N_NODESC * HC;         // N*H
    float* agg    = m   + (size_t)N_NODESC * HC;         // N*H
    float* pooled = agg + (size_t)N_NODESC * HC;         // G*H
    float* counts = pooled + (size_t)N_GRAPHSC * HC;     // G  (contiguous with pooled)
    float* w_inT  = counts + N_GRAPHSC;                  // H * D_IN
    float* ggcT   = w_inT + HC * D_INC;                  // 3 * H * H
    float* fc1T   = ggcT + 3 * HC * HC;                  // H * H
    float* fc2T   = fc1T + HC * HC;                      // H * 64

    // Pre-transpose small weights so WMMA B fragments load as one v2f each.
    k_tr<<<(D_INC * HC + 255) / 256, 256, 0, stream>>>(w_in, w_inT, D_INC, HC);
    for (int i = 0; i < 3; ++i)
        k_tr<<<(HC * HC + 255) / 256, 256, 0, stream>>>(ggc + (size_t)i * HC * HC,
                                                        ggcT + (size_t)i * HC * HC, HC, HC);
    k_tr<<<(HC * HC + 255) / 256, 256, 0, stream>>>(fc1w, fc1T, HC, HC);
    k_tr<<<(HC * 64 + 255) / 256, 256, 0, stream>>>(fc2w, fc2T, HC, 64);

    // h = relu(bn0(x @ w_in + b_in))
    k_in_gemm<<<N_NODESC / 16, 256, 0, stream>>>(x, w_inT, b_in, bn0g, bn0b, bn0m, bn0v, h);

    for (int i = 0; i < 3; ++i) {
        const int n4 = N_NODESC * HC / 4;
        k_zero4<<<(n4 + 255) / 256, 256, 0, stream>>>((float4*)agg, n4);
        k_mm<<<N_NODESC / 16, 256, 0, stream>>>(h, ggcT + (size_t)i * HC * HC, m);
        k_scatter<<<N_EDGESC / 8, 256, 0, stream>>>(m, ei, agg);
        k_gru<<<N_NODESC / 16, 256, 0, stream>>>(h, agg, wih, whh, bih, bhh);
    }

    const int p4 = (N_GRAPHSC * HC + N_GRAPHSC) / 4;   // pooled + counts together
    k_zero4<<<(p4 + 255) / 256, 256, 0, stream>>>((float4*)pooled, p4);
    k_pool<<<N_NODESC / 8, 256, 0, stream>>>(h, batch, bn1g, bn1b, bn1m, bn1v, pooled, counts);

    k_head<<<N_GRAPHSC / 16, 256, 0, stream>>>(pooled, counts,
                                               fc1T, fc1b, bn2g, bn2b, bn2m, bn2v,
                                               fc2T, fc2b, bn3g, bn3b, bn3m, bn3v,
                                               fc3w, fc3b, (float*)d_out);
}